// LinearMultiHeadAttention_15092515078295
// MI455X (gfx1250) — compile-verified
//
#include <hip/hip_runtime.h>
#include <math.h>

typedef float v2f __attribute__((ext_vector_type(2)));
typedef float v4f __attribute__((ext_vector_type(4)));
typedef float v8f __attribute__((ext_vector_type(8)));
typedef int   v4i __attribute__((ext_vector_type(4)));

#define B_ 8
#define N_ 2048
#define D_ 1024
#define H_ 16
#define DH_ 64
#define KR_ 128

// ---------------------------------------------------------------------------
// gfx1250 async global->LDS (GLOBAL_LOAD_ASYNC_TO_LDS_B128, ASYNCcnt-tracked)
// ---------------------------------------------------------------------------
typedef __attribute__((address_space(1))) v4i glob_v4i;
typedef __attribute__((address_space(3))) v4i lds_v4i;
typedef __attribute__((address_space(1))) void glob_void;
typedef __attribute__((address_space(3))) void lds_void;

#if __has_builtin(__builtin_amdgcn_global_load_async_to_lds_b128)
#define USE_ASYNC_LDS 1
#endif

__device__ __forceinline__ void stage16(const float* gp, float* lp)
{
#if defined(USE_ASYNC_LDS)
    __builtin_amdgcn_global_load_async_to_lds_b128(
        (glob_v4i*)(glob_void*)(void*)gp,
        (lds_v4i*)(lds_void*)(void*)lp, 0, 0);
#else
    *(v4f*)lp = *(const v4f*)gp;
#endif
}

__device__ __forceinline__ void wait_stage()
{
#if defined(USE_ASYNC_LDS)
#if __has_builtin(__builtin_amdgcn_s_wait_asynccnt)
    __builtin_amdgcn_s_wait_asynccnt(0);
#else
    asm volatile("s_wait_asynccnt 0x0" ::: "memory");
#endif
#endif
}

__device__ __forceinline__ void sched_fence()
{
#if __has_builtin(__builtin_amdgcn_sched_barrier)
    __builtin_amdgcn_sched_barrier(0);   // nothing may be scheduled across
#endif
}

// ---------------------------------------------------------------------------
// Generic batched tiled GEMM with fp32 WMMA (V_WMMA_F32_16X16X4_F32).
//   C[m][n] = alpha * sum_k A[m][k]*B[k][n]  (+ bias[n])
//   TA=0: A[m][k]=gA[m*lda+k]   TA=1: A[m][k]=gA[k*lda+m]
//   TB=0: B[k][n]=gB[k*ldb+n]   TB=1: B[k][n]=gB[n*ldb+k]
// Batch index z = b*Hc + h with separate (b,h) pointer strides.
// Block tile 128x64x16, 256 threads (8 wave32). Double-buffered LDS tiles:
// async-DMA of block t+1 overlaps WMMA compute of block t; one barrier/block.
// All problem dims are multiples of the tile dims -> no bounds checks.
// ---------------------------------------------------------------------------
template<bool TA, bool TB>
__global__ __launch_bounds__(256)
void gemm_wmma_f32(const float* __restrict__ gA, const float* __restrict__ gB,
                   float* __restrict__ gC, const float* __restrict__ bias,
                   int M, int N, int K, int lda, int ldb, int ldc,
                   long sAb, long sAh, long sBb, long sBh, long sCb, long sCh,
                   int Hc, float alpha)
{
    constexpr int BM = 128, BN = 64, BK = 16, SA = BK + 4;   // rows padded to 20
    __shared__ float As[2][BM * SA];   // As[buf][m][k]
    __shared__ float Bs[2][BN * SA];   // Bs[buf][n][k]

    const int bz = blockIdx.z;
    const int bb = bz / Hc, hh = bz % Hc;
    gA += bb * sAb + hh * sAh;
    gB += bb * sBb + hh * sBh;
    gC += bb * sCb + hh * sCh;

    const int mb   = blockIdx.y * BM;
    const int nb   = blockIdx.x * BN;
    const int tid  = threadIdx.x;
    const int wave = tid >> 5;         // 0..7
    const int lane = tid & 31;
    const int l15  = lane & 15;
    const int lhi  = lane >> 4;        // 0 or 1

    v8f acc[4];
    #pragma unroll
    for (int i = 0; i < 4; ++i) acc[i] = (v8f)0.0f;

    // stage one BK-deep tile pair into LDS buffer `buf` (known trip counts:
    // 2 vec4 transfers/thread for A, 1 for B -> no exec-mask guards)
    auto stage = [&](int buf, int kk) {
        float* Asb = As[buf];
        float* Bsb = Bs[buf];
        if (TA) {
            #pragma unroll
            for (int it = 0; it < 2; ++it) {
                int idx = tid + it * 256;
                int k = idx >> 5, m4 = (idx & 31) << 2;        // vec4 along m
                v4f t = *(const v4f*)&gA[(long)(kk + k) * lda + (mb + m4)];
                Asb[(m4 + 0) * SA + k] = t[0];
                Asb[(m4 + 1) * SA + k] = t[1];
                Asb[(m4 + 2) * SA + k] = t[2];
                Asb[(m4 + 3) * SA + k] = t[3];
            }
        } else {
            #pragma unroll
            for (int it = 0; it < 2; ++it) {
                int idx = tid + it * 256;
                int m = idx >> 2, kq = (idx & 3) << 2;         // vec4 along k
                stage16(&gA[(long)(mb + m) * lda + (kk + kq)], &Asb[m * SA + kq]);
            }
        }
        if (TB) {
            int n = tid >> 2, kq = (tid & 3) << 2;             // vec4 along k
            stage16(&gB[(long)(nb + n) * ldb + (kk + kq)], &Bsb[n * SA + kq]);
        } else {
            int k = tid >> 4, n4 = (tid & 15) << 2;            // vec4 along n
            v4f t = *(const v4f*)&gB[(long)(kk + k) * ldb + (nb + n4)];
            Bsb[(n4 + 0) * SA + k] = t[0];
            Bsb[(n4 + 1) * SA + k] = t[1];
            Bsb[(n4 + 2) * SA + k] = t[2];
            Bsb[(n4 + 3) * SA + k] = t[3];
        }
    };

    // prologue: fill buffer 0
    stage(0, 0);
    wait_stage();
    __syncthreads();

    const int nkb = K / BK;
    const int mrow = wave * 16 + l15;
    for (int t = 0; t < nkb; ++t) {
        const int p = t & 1;
        if (t + 1 < nkb) stage(p ^ 1, (t + 1) * BK);   // async: overlaps compute

        const float* Asb = As[p];
        const float* Bsb = Bs[p];
        #pragma unroll
        for (int k4 = 0; k4 < BK; k4 += 4) {
            const int kb = k4 + (lhi << 1);   // lanes 0-15 -> K0,K1 ; 16-31 -> K2,K3
            v2f a = *reinterpret_cast<const v2f*>(&Asb[mrow * SA + kb]);
            #pragma unroll
            for (int nt = 0; nt < 4; ++nt) {
                v2f b = *reinterpret_cast<const v2f*>(&Bsb[(nt * 16 + l15) * SA + kb]);
                acc[nt] = __builtin_amdgcn_wmma_f32_16x16x4_f32(
                    false, a, false, b, (short)0, acc[nt], false, false);
            }
        }
        sched_fence();      // keep the WMMAs ABOVE the async-wait + barrier
        wait_stage();       // next-buffer DMA done (hidden behind the WMMAs)
        __syncthreads();
    }

    // ---- store: lane holds rows (8*lhi + r), col l15, per n-tile ----
    const int rbase = mb + wave * 16 + (lhi << 3);
    #pragma unroll
    for (int nt = 0; nt < 4; ++nt) {
        const int n  = nb + nt * 16 + l15;
        const float bv = bias ? bias[n] : 0.0f;
        #pragma unroll
        for (int r = 0; r < 8; ++r) {
            gC[(long)(rbase + r) * ldc + n] = acc[nt][r] * alpha + bv;
        }
    }
}

// ---------------------------------------------------------------------------
// In-place row softmax, row length 2048, one 256-thread block per row.
// ---------------------------------------------------------------------------
__global__ __launch_bounds__(256)
void softmax_rows_2048(float* __restrict__ data)
{
    __shared__ float red[256];
    v4f* row4 = (v4f*)(data + (long)blockIdx.x * 2048);
    const int tid = threadIdx.x;

    v4f v[2];
    float mx = -INFINITY;
    #pragma unroll
    for (int i = 0; i < 2; ++i) {
        v[i] = row4[tid + i * 256];
        #pragma unroll
        for (int j = 0; j < 4; ++j) mx = fmaxf(mx, v[i][j]);
    }
    red[tid] = mx; __syncthreads();
    for (int s = 128; s > 0; s >>= 1) {
        if (tid < s) red[tid] = fmaxf(red[tid], red[tid + s]);
        __syncthreads();
    }
    mx = red[0]; __syncthreads();

    float sum = 0.0f;
    #pragma unroll
    for (int i = 0; i < 2; ++i)
        #pragma unroll
        for (int j = 0; j < 4; ++j) { v[i][j] = __expf(v[i][j] - mx); sum += v[i][j]; }
    red[tid] = sum; __syncthreads();
    for (int s = 128; s > 0; s >>= 1) {
        if (tid < s) red[tid] += red[tid + s];
        __syncthreads();
    }
    const float inv = 1.0f / red[0];
    #pragma unroll
    for (int i = 0; i < 2; ++i) {
        v4f o;
        #pragma unroll
        for (int j = 0; j < 4; ++j) o[j] = v[i][j] * inv;
        row4[tid + i * 256] = o;
    }
}

// ---------------------------------------------------------------------------
extern "C" void kernel_launch(void* const* d_in, const int* in_sizes, int n_in,
                              void* d_out, int out_size, void* d_ws, size_t ws_size,
                              hipStream_t stream)
{
    const float* q      = (const float*)d_in[0];
    const float* k      = (const float*)d_in[1];
    const float* v      = (const float*)d_in[2];
    const float* Wq     = (const float*)d_in[3];
    const float* bq     = (const float*)d_in[4];
    const float* Wk     = (const float*)d_in[5];
    const float* bk     = (const float*)d_in[6];
    const float* Wv     = (const float*)d_in[7];
    const float* bv     = (const float*)d_in[8];
    const float* projk  = (const float*)d_in[9];
    const float* projv  = (const float*)d_in[10];
    const float* Wo     = (const float*)d_in[11];
    const float* bo     = (const float*)d_in[12];
    float* out = (float*)d_out;

    // workspace layout (floats); scores aliases Kp+Vp, Ocat aliases Qp
    float* ws = (float*)d_ws;
    const long QP_N = (long)B_ * N_ * D_;            // 16,777,216
    float* Qp     = ws;                              // [B*N, D]
    float* Kp     = ws + QP_N;                       // [B*N, D]
    float* Vp     = ws + 2 * QP_N;                   // [B*N, D]
    float* pk     = ws + 3 * QP_N;                   // [B, KR, D]
    float* pv     = pk + (long)B_ * KR_ * D_;        // [B, KR, D]
    float* scores = Kp;                              // [B, H, KR, N] (overlays Kp+Vp)
    float* Ocat   = Qp;                              // [B, N, D]    (overlays Qp)

    const int MT = B_ * N_;  // 16384

    // 1-3) Q/K/V projections: Y = X @ W^T + b
    dim3 gProj(D_ / 64, MT / 128, 1);
    gemm_wmma_f32<false, true><<<gProj, 256, 0, stream>>>(
        k, Wk, Kp, bk, MT, D_, D_, D_, D_, D_, 0, 0, 0, 0, 0, 0, 1, 1.0f);
    gemm_wmma_f32<false, true><<<gProj, 256, 0, stream>>>(
        q, Wq, Qp, bq, MT, D_, D_, D_, D_, D_, 0, 0, 0, 0, 0, 0, 1, 1.0f);
    gemm_wmma_f32<false, true><<<gProj, 256, 0, stream>>>(
        v, Wv, Vp, bv, MT, D_, D_, D_, D_, D_, 0, 0, 0, 0, 0, 0, 1, 1.0f);

    // 4-5) low-rank projections: pk[b,kr,d] = sum_n projk[n,kr] * Kp[b,n,d]
    dim3 gPk(D_ / 64, KR_ / 128, B_);
    gemm_wmma_f32<true, false><<<gPk, 256, 0, stream>>>(
        projk, Kp, pk, nullptr, KR_, D_, N_, KR_, D_, D_,
        0, 0, (long)N_ * D_, 0, (long)KR_ * D_, 0, 1, 1.0f);
    gemm_wmma_f32<true, false><<<gPk, 256, 0, stream>>>(
        projv, Vp, pv, nullptr, KR_, D_, N_, KR_, D_, D_,
        0, 0, (long)N_ * D_, 0, (long)KR_ * D_, 0, 1, 1.0f);

    // 6) scores[b,h,kr,n] = (1/8) sum_d pk[b,h,kr,d] * qp[b,h,n,d]
    //    qp view: Qp_flat[b*N*D + h*N_*DH_ + n*64 + d]  -> TB with ldb=64
    dim3 gSc(N_ / 64, KR_ / 128, B_ * H_);
    gemm_wmma_f32<false, true><<<gSc, 256, 0, stream>>>(
        pk, Qp, scores, nullptr, KR_, N_, DH_, D_, DH_, N_,
        (long)KR_ * D_, DH_,                       // A strides (b, h)
        (long)N_ * D_, (long)N_ * DH_,             // B strides (b, h)
        (long)H_ * KR_ * N_, (long)KR_ * N_,       // C strides (b, h)
        H_, 0.125f);

    // 7) softmax over N per (b,h,kr) row
    softmax_rows_2048<<<B_ * H_ * KR_, 256, 0, stream>>>(scores);

    // 8) Ocat[b,n,h*64+d] = sum_kr attn[b,h,kr,n] * pv[b,h,kr,d]
    dim3 gOut(DH_ / 64, N_ / 128, B_ * H_);
    gemm_wmma_f32<true, false><<<gOut, 256, 0, stream>>>(
        scores, pv, Ocat, nullptr, N_, DH_, KR_, N_, D_, D_,
        (long)H_ * KR_ * N_, (long)KR_ * N_,       // A strides (b, h)
        (long)KR_ * D_, DH_,                       // B strides (b, h)
        (long)N_ * D_, DH_,                        // C strides (b, h)
        H_, 1.0f);

    // 9) final projection: out = Ocat @ Wo^T + bo
    gemm_wmma_f32<false, true><<<gProj, 256, 0, stream>>>(
        Ocat, Wo, out, bo, MT, D_, D_, D_, D_, D_, 0, 0, 0, 0, 0, 0, 1, 1.0f);
}